// Attention_57956288692508
// MI455X (gfx1250) — compile-verified
//
#include <hip/hip_runtime.h>
#include <math.h>

// ---------------------------------------------------------------------------
// Gaussian-windowed local attention (Luong-style predicted position).
//   B=16, S=4096, H=1024, WINDOW=64 (sigma=32).
// Memory-bound: source = 256 MB fp32. Single-pass online softmax reads it once
// (~11 us at 23.3 TB/s) instead of twice. WMMA used for the t@W_p GEMM.
// ---------------------------------------------------------------------------

#define BATCH 16
#define SEQ   4096
#define HID   1024

typedef __attribute__((ext_vector_type(2))) float v2f;
typedef __attribute__((ext_vector_type(8))) float v8f;

__device__ __forceinline__ float dot4(const float4& a, const float4& b) {
    return a.x * b.x + a.y * b.y + a.z * b.z + a.w * b.w;
}

// ---------------------------------------------------------------------------
// Kernel 1: p[b] = sigmoid( tanh(t_b @ W_p + b_p) . v_p + b_v ) * SEQ
// One block, 1024 threads = 32 waves. Wave w computes N-tiles {w, w+32}
// (16 cols each) of the 16x1024 GEMM via V_WMMA_F32_16X16X4_F32 over K=1024.
// ISA 7.12.2 fp32 layouts:
//   A 16x4 : lane<16 -> (M=lane, K=k0,k0+1) ; lane>=16 -> (M=lane-16, K=k0+2,k0+3)
//   B 4x16 : V0: K=k0 (lane<16) / k0+2 (lane>=16), N=lane%16 ; V1: K=k0+1 / k0+3
//   C 16x16: VGPR r: M=r (lane<16) / r+8 (lane>=16), N=lane%16
// ---------------------------------------------------------------------------
__global__ void __launch_bounds__(1024)
compute_p_kernel(const float* __restrict__ tgt,   // (16,1024)
                 const float* __restrict__ Wp,    // (1024,1024) row-major [h][j]
                 const float* __restrict__ bp,    // (1024)
                 const float* __restrict__ vp,    // (1024)
                 const float* __restrict__ bv,    // scalar
                 float* __restrict__ ws_p)        // (16) out
{
    __shared__ float lds_part[32 * 16];

    const int tid  = threadIdx.x;
    const int wave = tid >> 5;
    const int lane = tid & 31;
    const int half = lane >> 4;   // 0: lanes 0-15, 1: lanes 16-31
    const int m16  = lane & 15;

    float pr[8];
#pragma unroll
    for (int r = 0; r < 8; ++r) pr[r] = 0.0f;

#pragma unroll
    for (int t = 0; t < 2; ++t) {
        const int tile = wave + 32 * t;     // 0..63
        const int n0   = tile * 16;

        v8f c = {};
        for (int k0 = 0; k0 < HID; k0 += 4) {
            // A fragment: t[m16, k0 + 2*half .. +1]
            v2f a = *(const v2f*)(tgt + m16 * HID + k0 + 2 * half);
            // B fragment: W_p[k, n0+m16]
            v2f bm;
            bm.x = Wp[(size_t)(k0 + 2 * half) * HID + n0 + m16];
            bm.y = Wp[(size_t)(k0 + 2 * half + 1) * HID + n0 + m16];
            c = __builtin_amdgcn_wmma_f32_16x16x4_f32(
                    false, a, false, bm, (short)0, c, false, false);
        }

        // Epilogue: hp = tanh(c + b_p[n]); contrib = hp * v_p[n]; reduce over N.
        const int n = n0 + m16;
        const float bpn = bp[n];
        const float vpn = vp[n];
#pragma unroll
        for (int r = 0; r < 8; ++r) {
            float hp = tanhf(c[r] + bpn);
            float contrib = hp * vpn;
            // sum across the 16 lanes of this half (masks 1..8 stay in-half)
#pragma unroll
            for (int mask = 8; mask >= 1; mask >>= 1)
                contrib += __shfl_xor(contrib, mask, 32);
            pr[r] += contrib;   // partial for batch M = r + 8*half over this N-tile
        }
    }

    // lanes 0 and 16 hold the half-sums; deposit per-wave partials.
    if (m16 == 0) {
#pragma unroll
        for (int r = 0; r < 8; ++r)
            lds_part[wave * 16 + (r + 8 * half)] = pr[r];
    }
    __syncthreads();

    if (tid < BATCH) {
        float acc = 0.0f;
        for (int w = 0; w < 32; ++w) acc += lds_part[w * 16 + tid];
        acc += bv[0];
        float sig = 1.0f / (1.0f + __expf(-acc));
        ws_p[tid] = sig * (float)SEQ;
    }
}

// ---------------------------------------------------------------------------
// Kernel 2: single-pass online-softmax attention.
// Grid (32 chunks, 16 batches) x 256 threads (8 waves). Each wave owns 16 rows
// (s = chunk*128 + wave + 8*r). A full 1024-float row lives in 8 float4 regs;
// the dot with t_b (also in regs) is reduced with wave32 shfl_xor. Running
// (m, l, ctx[1024]) per wave with max-rescaling; the Gaussian window enters the
// numerator only (softmax normalizes over raw scores, per the reference).
// Partials go to scratch; kernel 3 merges them. Source is read exactly once.
// ---------------------------------------------------------------------------
__global__ void __launch_bounds__(256)
flash_attn_kernel(const float* __restrict__ src,   // (16,4096,1024)
                  const float* __restrict__ tgt,   // (16,1024)
                  const float* __restrict__ ws_p,  // (16)
                  float* __restrict__ ws_ctx,      // (16,256,1024)
                  float* __restrict__ ws_m,        // (16,256)
                  float* __restrict__ ws_l)        // (16,256)
{
    const int tid   = threadIdx.x;
    const int lane  = tid & 31;
    const int wave  = tid >> 5;
    const int chunk = blockIdx.x;   // 0..31
    const int b     = blockIdx.y;   // 0..15

    const float4* srcb = (const float4*)(src + (size_t)b * SEQ * HID);
    const float4* tgt4 = (const float4*)(tgt + (size_t)b * HID);

    // t_b slice for this lane: h = lane*4 + i*128
    float4 tv[8];
#pragma unroll
    for (int i = 0; i < 8; ++i) tv[i] = tgt4[lane + i * 32];

    const float p = ws_p[b];

    float m = -1e30f, l = 0.0f;
    float4 ctx[8];
#pragma unroll
    for (int i = 0; i < 8; ++i) ctx[i] = make_float4(0.f, 0.f, 0.f, 0.f);

    for (int r = 0; r < 16; ++r) {
        const int s = chunk * 128 + wave + r * 8;
        const float4* row = srcb + (size_t)s * (HID / 4);

        if (r + 1 < 16) // prefetch next row (gfx1250 global_prefetch_b8)
            __builtin_prefetch(srcb + (size_t)(s + 8) * (HID / 4) + lane, 0, 1);

        float4 rv[8];
#pragma unroll
        for (int i = 0; i < 8; ++i) rv[i] = row[lane + i * 32];

        float d = 0.0f;
#pragma unroll
        for (int i = 0; i < 8; ++i) d += dot4(rv[i], tv[i]);
#pragma unroll
        for (int off = 16; off >= 1; off >>= 1)
            d += __shfl_xor(d, off, 32);

        const float score = d * 0.03125f;   // 1/sqrt(1024)

        if (score > m) {                    // wave-uniform branch
            const float sc = __expf(m - score);
            l *= sc;
#pragma unroll
            for (int i = 0; i < 8; ++i) {
                ctx[i].x *= sc; ctx[i].y *= sc; ctx[i].z *= sc; ctx[i].w *= sc;
            }
            m = score;
        }
        const float w = __expf(score - m);
        l += w;

        const float dp = (float)s - p;
        const float g  = __expf(-dp * dp * (1.0f / 2048.0f)); // 2*sigma^2, sigma=32
        const float wg = w * g;
#pragma unroll
        for (int i = 0; i < 8; ++i) {
            ctx[i].x += wg * rv[i].x; ctx[i].y += wg * rv[i].y;
            ctx[i].z += wg * rv[i].z; ctx[i].w += wg * rv[i].w;
        }
    }

    const int c = chunk * 8 + wave;         // 0..255 per batch
    if (lane == 0) {
        ws_m[b * 256 + c] = m;
        ws_l[b * 256 + c] = l;
    }
    float4* cp = (float4*)(ws_ctx + ((size_t)b * 256 + c) * HID);
#pragma unroll
    for (int i = 0; i < 8; ++i) cp[lane + i * 32] = ctx[i];
}

// ---------------------------------------------------------------------------
// Kernel 3: merge 256 per-wave partials per batch.
//   M = max m_c ; denom = sum l_c * exp(m_c - M)
//   out[b,h] = (sum_c ctx_c[h] * exp(m_c - M)) / denom
// Grid 16 x 256 threads; thread -> 4 h values (float4). 16 MB read, trivial.
// ---------------------------------------------------------------------------
__global__ void __launch_bounds__(256)
combine_kernel(const float* __restrict__ ws_ctx,
               const float* __restrict__ ws_m,
               const float* __restrict__ ws_l,
               float* __restrict__ out)          // (16,1024)
{
    __shared__ float coef[256];
    __shared__ float red[256];

    const int b   = blockIdx.x;
    const int tid = threadIdx.x;

    const float mc = ws_m[b * 256 + tid];
    const float lc = ws_l[b * 256 + tid];

    red[tid] = mc;
    __syncthreads();
    for (int s = 128; s > 0; s >>= 1) {
        if (tid < s) red[tid] = fmaxf(red[tid], red[tid + s]);
        __syncthreads();
    }
    const float M = red[0];
    __syncthreads();

    const float co = __expf(mc - M);
    coef[tid] = co;
    red[tid]  = lc * co;
    __syncthreads();
    for (int s = 128; s > 0; s >>= 1) {
        if (tid < s) red[tid] += red[tid + s];
        __syncthreads();
    }
    const float inv = 1.0f / red[0];
    __syncthreads();

    const float4* base = (const float4*)(ws_ctx + (size_t)b * 256 * HID) + tid;
    float4 acc = make_float4(0.f, 0.f, 0.f, 0.f);
    for (int c = 0; c < 256; ++c) {
        const float4 v = base[(size_t)c * (HID / 4)];
        const float w = coef[c];
        acc.x += w * v.x; acc.y += w * v.y; acc.z += w * v.z; acc.w += w * v.w;
    }
    float4* o4 = (float4*)(out + (size_t)b * HID) + tid;
    *o4 = make_float4(acc.x * inv, acc.y * inv, acc.z * inv, acc.w * inv);
}

// ---------------------------------------------------------------------------
extern "C" void kernel_launch(void* const* d_in, const int* in_sizes, int n_in,
                              void* d_out, int out_size, void* d_ws, size_t ws_size,
                              hipStream_t stream) {
    const float* src = (const float*)d_in[0];  // (16,4096,1024)
    const float* tgt = (const float*)d_in[1];  // (16,1024)
    const float* Wp  = (const float*)d_in[2];  // (1024,1024)
    const float* bp  = (const float*)d_in[3];  // (1024)
    const float* vp  = (const float*)d_in[4];  // (1024)
    const float* bv  = (const float*)d_in[5];  // scalar
    float* out = (float*)d_out;                // (16,1024)

    // Scratch layout: ctx partials (16 MB), then m, l, p. All written before read.
    float* ws_ctx = (float*)d_ws;                       // 16*256*1024 floats
    float* ws_m   = ws_ctx + (size_t)BATCH * 256 * HID; // 16*256
    float* ws_l   = ws_m + BATCH * 256;                 // 16*256
    float* ws_p   = ws_l + BATCH * 256;                 // 16

    compute_p_kernel<<<1, 1024, 0, stream>>>(tgt, Wp, bp, vp, bv, ws_p);
    flash_attn_kernel<<<dim3(32, BATCH), 256, 0, stream>>>(src, tgt, ws_p,
                                                           ws_ctx, ws_m, ws_l);
    combine_kernel<<<BATCH, 256, 0, stream>>>(ws_ctx, ws_m, ws_l, out);
}